// MNL_loss_37168646980396
// MI455X (gfx1250) — compile-verified
//
#include <hip/hip_runtime.h>
#include <math.h>

typedef __attribute__((ext_vector_type(2))) float v2f;
typedef __attribute__((ext_vector_type(8))) float v8f;

#define COLS            1024
#define ROWS_PER_TILE   16
#define WAVES_PER_BLOCK 8

// One wave per 16-row tile.
// Pass A: per-row max (lane L covers half of row L&15; shfl_xor(16) combines).
// Pass B: three v_wmma_f32_16x16x4_f32 accumulator chains compute, per row:
//         sum(exp(x-max)), sum(x*y), sum(y)   via  D = A(16x4) * ones(4x16) + C.
// A-operand f32 layout: M = lane&15; lanes 0-15 supply K=0,1 ; lanes 16-31 K=2,3.
__global__ __launch_bounds__(256) void mnl_tile_kernel(
    const float* __restrict__ x, const float* __restrict__ y,
    const float* __restrict__ u, float* __restrict__ partials)
{
    __shared__ float exch[WAVES_PER_BLOCK][3][16];
    __shared__ float wpart[WAVES_PER_BLOCK];

    const int lane = threadIdx.x & 31;
    const int wave = threadIdx.x >> 5;
    const int tile = blockIdx.x * WAVES_PER_BLOCK + wave;
    const int row  = tile * ROWS_PER_TILE + (lane & 15);
    const int half = lane >> 4;          // 0: lanes 0-15, 1: lanes 16-31

    const float* rowx = x + (size_t)row * COLS;
    const float* rowy = y + (size_t)row * COLS;

    // ---------------- Pass A: row max ----------------
    float m = -INFINITY;
    {
        const float4* px4 = (const float4*)(rowx + half * (COLS / 2));
        #pragma unroll 4
        for (int c = 0; c < COLS / 8; ++c) {
            float4 v = px4[c];
            m = fmaxf(m, fmaxf(fmaxf(v.x, v.y), fmaxf(v.z, v.w)));
        }
    }
    m = fmaxf(m, __shfl_xor(m, 16, 32));   // full-row max in every lane of the pair

    // ---------------- Pass B: WMMA row reductions ----------------
    const float* pxb = rowx + half * 2;    // K=0,1 for lanes 0-15; K=2,3 for 16-31
    const float* pyb = rowy + half * 2;

    v8f accE = {0.f,0.f,0.f,0.f,0.f,0.f,0.f,0.f};   // sum exp(x - m)
    v8f accD = accE;                                 // sum x*y
    v8f accY = accE;                                 // sum y
    v2f ones; ones.x = 1.0f; ones.y = 1.0f;          // B = all-ones 4x16

    for (int k = 0; k < COLS; k += 16) {
        #pragma unroll
        for (int j = 0; j < 4; ++j) {
            float2 xv = *(const float2*)(pxb + k + 4 * j);
            float2 yv = *(const float2*)(pyb + k + 4 * j);
            v2f aE, aD, aY;
            aE.x = __expf(xv.x - m);  aE.y = __expf(xv.y - m);
            aD.x = xv.x * yv.x;       aD.y = xv.y * yv.y;
            aY.x = yv.x;              aY.y = yv.y;
            accE = __builtin_amdgcn_wmma_f32_16x16x4_f32(
                       false, aE, false, ones, (short)0, accE, false, false);
            accD = __builtin_amdgcn_wmma_f32_16x16x4_f32(
                       false, aD, false, ones, (short)0, accD, false, false);
            accY = __builtin_amdgcn_wmma_f32_16x16x4_f32(
                       false, aY, false, ones, (short)0, accY, false, false);
        }
    }

    // D layout: VGPR v -> row v (lanes 0-15) / row v+8 (lanes 16-31), value
    // replicated across all N columns. Exchange through LDS to map row r -> lane r.
    if (lane == 0) {
        #pragma unroll
        for (int v = 0; v < 8; ++v) {
            exch[wave][0][v] = accE[v];
            exch[wave][1][v] = accD[v];
            exch[wave][2][v] = accY[v];
        }
    }
    if (lane == 16) {
        #pragma unroll
        for (int v = 0; v < 8; ++v) {
            exch[wave][0][8 + v] = accE[v];
            exch[wave][1][8 + v] = accD[v];
            exch[wave][2][8 + v] = accY[v];
        }
    }
    __syncthreads();

    const float u0 = u[0];
    float loss = 0.0f;
    if (lane < 16) {                        // lane r owns row r; m is that row's max
        float es = exch[wave][0][lane];
        float dt = exch[wave][1][lane];
        float ys = exch[wave][2][lane];
        float lse     = m + __logf(es);                         // logsumexp(x_row)
        float hi      = fmaxf(u0, lse);
        float lo      = fminf(u0, lse);
        float log_den = hi + log1pf(__expf(lo - hi));           // logaddexp(u0, lse)
        float s       = (ys > 0.0f) ? dt : u0;
        loss          = s - log_den;                            // / ln10 applied later
    }
    #pragma unroll
    for (int off = 16; off >= 1; off >>= 1)
        loss += __shfl_xor(loss, off, 32);
    if (lane == 0) wpart[wave] = loss;
    __syncthreads();

    if (threadIdx.x == 0) {
        float t = 0.0f;
        #pragma unroll
        for (int w = 0; w < WAVES_PER_BLOCK; ++w) t += wpart[w];
        partials[blockIdx.x] = t;           // deterministic per-block partial
    }
}

// Deterministic fixed-tree reduction of block partials; applies -1/(B*ln10).
__global__ __launch_bounds__(256) void mnl_finalize_kernel(
    const float* __restrict__ partials, int n, float* __restrict__ out, float scale)
{
    __shared__ float red[256];
    float t = 0.0f;
    for (int i = threadIdx.x; i < n; i += 256) t += partials[i];
    red[threadIdx.x] = t;
    __syncthreads();
    #pragma unroll
    for (int s = 128; s > 0; s >>= 1) {
        if (threadIdx.x < s) red[threadIdx.x] += red[threadIdx.x + s];
        __syncthreads();
    }
    if (threadIdx.x == 0) out[0] = red[0] * scale;
}

extern "C" void kernel_launch(void* const* d_in, const int* in_sizes, int n_in,
                              void* d_out, int out_size, void* d_ws, size_t ws_size,
                              hipStream_t stream)
{
    const float* x = (const float*)d_in[0];
    const float* y = (const float*)d_in[1];
    const float* u = (const float*)d_in[2];
    float* out = (float*)d_out;
    float* ws  = (float*)d_ws;

    const int Bn     = in_sizes[0] / COLS;                 // 65536
    const int tiles  = Bn / ROWS_PER_TILE;                 // 4096
    const int blocks = tiles / WAVES_PER_BLOCK;            // 512

    mnl_tile_kernel<<<blocks, 256, 0, stream>>>(x, y, u, ws);

    const float ln10  = 2.302585092994046f;
    const float scale = -1.0f / ((float)Bn * ln10);
    mnl_finalize_kernel<<<1, 256, 0, stream>>>(ws, blocks, out, scale);
}